// QuantumGeneratorImported_62740882260820
// MI455X (gfx1250) — compile-verified
//
#include <hip/hip_runtime.h>

#define NQ      14
#define DIM     16384      // 2^14 statevector
#define PIX     2048       // pixels per patch
#define NLAYERS 6
#define NSUB    2
#define TPB     256        // 8 wave32s

typedef __attribute__((ext_vector_type(2))) float v2f;
typedef __attribute__((ext_vector_type(8))) float v8f;

// CNOT-ring source permutation: psi'[y] = psi[perm(y)];
// each step is linear over GF(2), so perm(a^b) = perm(a)^perm(b).
__host__ __device__ constexpr int cnot_perm(int y) {
  int t = y;
  for (int w = NQ - 1; w >= 0; --w) {
    const int cb = (NQ - 1) - w;
    const int tb = (NQ - 1) - ((w + 1) % NQ);
    t ^= ((t >> cb) & 1) << tb;
  }
  return t;
}

// perm(j << 10) for j = 0..15 : compile-time constants (group-0 row offsets).
struct PermJ { int v[16]; };
__host__ __device__ constexpr PermJ make_permJ() {
  PermJ r{};
  for (int j = 0; j < 16; ++j) r.v[j] = cnot_perm(j << 10);
  return r;
}
constexpr PermJ kPermJ = make_permJ();

// Entry (m,j) of R(a0) (x) R(a1) (x) R(a2) (x) R(a3); wire k <-> bit (3-k).
__device__ __forceinline__ float uentry(int m, int j, const float c[4], const float s[4]) {
  float u = 1.0f;
#pragma unroll
  for (int k = 0; k < 4; ++k) {
    const int mb = (m >> (3 - k)) & 1;
    const int jb = (j >> (3 - k)) & 1;
    u *= (mb == jb) ? c[k] : (mb ? s[k] : -s[k]);   // RY: [[c,-s],[s,c]]
  }
  return u;
}

// Apply 16x16 unitary U (Kron of 4 RYs) to state viewed as (Q,16,P), P=1<<SHIFTP.
// PERMUTE (SHIFTP==10 only): B-loads read src[perm(addr)] via the GF(2)-linear
// decomposition, folding the previous layer's CNOT ring into this gate pass.
template <int SHIFTP, bool PERMUTE>
__device__ __forceinline__ void apply_group(float* dst, const float* src,
                                            float a0, float a1, float a2, float a3,
                                            int wave, int lane) {
  float c[4], s[4];
  const float a[4] = {a0, a1, a2, a3};
#pragma unroll
  for (int k = 0; k < 4; ++k) { c[k] = __cosf(0.5f * a[k]); s[k] = __sinf(0.5f * a[k]); }

  const int m  = lane & 15;
  const int hi = lane >> 4;
  // A fragments (16x4 slices of U): VGPR0 -> K = 4kk + {0|2}, VGPR1 -> K = 4kk + {1|3}
  v2f A[4];
#pragma unroll
  for (int kk = 0; kk < 4; ++kk) {
    A[kk].x = uentry(m, kk * 4 + (hi ? 2 : 0), c, s);
    A[kk].y = uentry(m, kk * 4 + (hi ? 3 : 1), c, s);
  }

  const int r0 = hi ? 2 : 0;
  const int r1 = hi ? 3 : 1;

  // 1024 columns -> 64 tiles of N=16; waves stride tiles (uniform, EXEC all-1s).
  for (int t = wave; t < 64; t += 8) {
    const int col  = t * 16 + m;
    const int q    = col >> SHIFTP;
    const int p    = col & ((1 << SHIFTP) - 1);
    const int base = (q << (SHIFTP + 4)) + p;

    v2f B[4];
    if (PERMUTE) {
      const int pp = cnot_perm(col);             // one 14-step chain per column
#pragma unroll
      for (int kk = 0; kk < 4; ++kk) {
        B[kk].x = src[pp ^ kPermJ.v[kk * 4 + r0]];
        B[kk].y = src[pp ^ kPermJ.v[kk * 4 + r1]];
      }
    } else {
#pragma unroll
      for (int kk = 0; kk < 4; ++kk) {
        B[kk].x = src[base + ((kk * 4 + r0) << SHIFTP)];
        B[kk].y = src[base + ((kk * 4 + r1) << SHIFTP)];
      }
    }

    v8f acc = {0.f, 0.f, 0.f, 0.f, 0.f, 0.f, 0.f, 0.f};
#pragma unroll
    for (int kk = 0; kk < 4; ++kk) {
      acc = __builtin_amdgcn_wmma_f32_16x16x4_f32(
          false, A[kk], false, B[kk], (short)0, acc, false, false);
    }

#pragma unroll
    for (int r = 0; r < 8; ++r) {
      dst[base + ((r + hi * 8) << SHIFTP)] = acc[r];
    }
  }
}

__global__ void __launch_bounds__(TPB)
qgen_kernel(const float* __restrict__ x, const float* __restrict__ params,
            float* __restrict__ out) {
  extern __shared__ float lds[];
  float* buf0 = lds;            // 16384 f32
  float* buf1 = lds + DIM;      // 16384 f32
  float* red  = lds + 2 * DIM;  // 256 f32 reduction scratch

  const int b    = blockIdx.x >> 1;
  const int g    = blockIdx.x & 1;
  const int tid  = threadIdx.x;
  const int wave = tid >> 5;
  const int lane = tid & 31;

  // |0...0>
  for (int i = tid; i < DIM; i += TPB) buf0[i] = (i == 0) ? 1.0f : 0.0f;
  __syncthreads();

  const float* xb = x + b * NQ;
  const float* pg = params + g * (NLAYERS * NQ);

  float* cur = buf0;
  float* nxt = buf1;

  // Encoding layer (no pending permutation).
  apply_group<10, false>(cur, cur, xb[0], xb[1], xb[2],  xb[3],  wave, lane); __syncthreads();
  apply_group<6,  false>(cur, cur, xb[4], xb[5], xb[6],  xb[7],  wave, lane); __syncthreads();
  apply_group<2,  false>(cur, cur, xb[8], xb[9], xb[10], xb[11], wave, lane); __syncthreads();
  apply_group<0,  false>(cur, cur, 0.0f,  0.0f,  xb[12], xb[13], wave, lane); __syncthreads();

  for (int l = 0; l < NLAYERS; ++l) {
    const float* pl = pg + l * NQ;
    if (l == 0) {
      // No permutation pending before layer 0's gates.
      apply_group<10, false>(cur, cur, pl[0], pl[1], pl[2], pl[3], wave, lane);
    } else {
      // Fold previous layer's CNOT ring into this group-0 pass (cur -> nxt).
      apply_group<10, true>(nxt, cur, pl[0], pl[1], pl[2], pl[3], wave, lane);
      float* tmp = cur; cur = nxt; nxt = tmp;
    }
    __syncthreads();
    apply_group<6, false>(cur, cur, pl[4], pl[5], pl[6],  pl[7],  wave, lane); __syncthreads();
    apply_group<2, false>(cur, cur, pl[8], pl[9], pl[10], pl[11], wave, lane); __syncthreads();
    apply_group<0, false>(cur, cur, 0.0f,  0.0f,  pl[12], pl[13], wave, lane); __syncthreads();
  }

  // Readout folds the final CNOT ring: logical i = (u<<10)|p, u<8 covers the
  // ancilla==0 half; post = p/max(p)*2 - 1 (sum normalization cancels).
  float mx = 0.0f;
  for (int p = tid; p < 1024; p += TPB) {
    const int pp = cnot_perm(p);
#pragma unroll
    for (int u = 0; u < 8; ++u) {
      const float v = cur[pp ^ kPermJ.v[u]];
      mx = fmaxf(mx, v * v);
    }
  }
  red[tid] = mx;
  __syncthreads();
#pragma unroll
  for (int off = TPB / 2; off > 0; off >>= 1) {
    if (tid < off) red[tid] = fmaxf(red[tid], red[tid + off]);
    __syncthreads();
  }
  const float scale = 2.0f / red[0];

  float* ob = out + (size_t)b * (NSUB * PIX) + (size_t)g * PIX;
  for (int p = tid; p < 1024; p += TPB) {
    const int pp = cnot_perm(p);
#pragma unroll
    for (int u = 0; u < 2; ++u) {
      const float v = cur[pp ^ kPermJ.v[u]];
      ob[u * 1024 + p] = v * v * scale - 1.0f;
    }
  }
}

extern "C" void kernel_launch(void* const* d_in, const int* in_sizes, int n_in,
                              void* d_out, int out_size, void* d_ws, size_t ws_size,
                              hipStream_t stream) {
  const float* x      = (const float*)d_in[0];   // (BATCH, 14) f32
  const float* params = (const float*)d_in[1];   // (2, 84)    f32
  float*       out    = (float*)d_out;           // (BATCH, 1, 64, 64) f32

  const int batch = in_sizes[0] / NQ;
  const size_t shmem = (size_t)(2 * DIM + TPB) * sizeof(float);  // 132 KB

  hipFuncSetAttribute(reinterpret_cast<const void*>(qgen_kernel),
                      hipFuncAttributeMaxDynamicSharedMemorySize, (int)shmem);

  qgen_kernel<<<dim3(batch * NSUB), dim3(TPB), shmem, stream>>>(x, params, out);
}